// CondMultiChannel2DCircularConv_84052509982879
// MI455X (gfx1250) — compile-verified
//
#include <hip/hip_runtime.h>
#include <hip/hip_bf16.h>
#include <math.h>

typedef __attribute__((ext_vector_type(2))) float v2f;
typedef __attribute__((ext_vector_type(8))) float v8f;

#define BATCH 4
#define CH    64
#define NSP   64
#define SPAT  (NSP*NSP)   // 4096
#define TAPS  9

#define WMMA_F32X4(A_, B_, C_) \
    __builtin_amdgcn_wmma_f32_16x16x4_f32(false, (A_), false, (B_), (short)0, (C_), false, false)

// ---------------------------------------------------------------------------
// Kernel 1: circular conv as 36 GEMMs (64co x 4096sp x 64ci) via f32 WMMA.
// One wave computes a 64co x 16sp output tile (4 accumulators of v8f).
// Grid: 1024 tiles / 8 waves per block = 128 blocks.
// ---------------------------------------------------------------------------
__global__ __launch_bounds__(256) void conv_wmma_kernel(
    const float* __restrict__ X,     // [B, C, N, N]
    const float* __restrict__ K,     // [B, C, C, 3, 3]
    const float* __restrict__ bias,  // [B, C]
    float* __restrict__ out)         // [B, C, N, N]
{
    const int lane  = threadIdx.x & 31;
    const int wave  = threadIdx.x >> 5;
    const int tile  = blockIdx.x * 8 + wave;      // 0..1023
    const int b     = tile >> 8;                  // tile / 256
    const int s0    = (tile & 255) << 4;          // 16 spatial positions
    const int y     = s0 >> 6;
    const int x0    = s0 & 63;
    const int lcol  = lane & 15;                  // B-frag N index / A-frag M index
    const int lhalf = lane >> 4;                  // selects K-pair {0,1} vs {2,3}

    const v8f zacc = {0.f,0.f,0.f,0.f,0.f,0.f,0.f,0.f};
    v8f acc[4];
    acc[0] = zacc; acc[1] = zacc; acc[2] = zacc; acc[3] = zacc;

    const float* Kb = K + b * CH * CH * TAPS;
    const float* Xb = X + b * CH * SPAT;

    for (int t = 0; t < TAPS; ++t) {
        const int ky = t / 3 - 1;
        const int kx = t % 3 - 1;
        const int yy = (y + ky) & 63;
        const int xx = (x0 + lcol + kx) & 63;     // per-lane circularly shifted column
        const float* Xrow = Xb + yy * 64 + xx;    // + ci*4096 selects channel

        #pragma unroll 4
        for (int k4 = 0; k4 < 16; ++k4) {
            const int ci0 = k4 * 4 + lhalf * 2;   // this lane-half's K indices {ci0, ci0+1}
            v2f bf;
            bf.x = Xrow[ci0 * SPAT];
            bf.y = Xrow[(ci0 + 1) * SPAT];

            #pragma unroll
            for (int m = 0; m < 4; ++m) {
                const int row = m * 16 + lcol;    // output channel co
                const float* Kr = Kb + (row * CH + ci0) * TAPS + t;
                v2f af;
                af.x = Kr[0]    + ((row == ci0     && t == 4) ? 1.0f : 0.0f);
                af.y = Kr[TAPS] + ((row == ci0 + 1 && t == 4) ? 1.0f : 0.0f);
                acc[m] = WMMA_F32X4(af, bf, acc[m]);
            }
        }
    }

    float* Ob = out + b * CH * SPAT;
    #pragma unroll
    for (int m = 0; m < 4; ++m) {
        #pragma unroll
        for (int r = 0; r < 8; ++r) {
            const int co = m * 16 + r + 8 * lhalf;   // C/D layout: M = vgpr + 8*(lane>=16)
            Ob[co * SPAT + s0 + lcol] = acc[m][r] + bias[b * CH + co];
        }
    }
}

// ---------------------------------------------------------------------------
// Kernel 2: per-frequency complex 64x64 log|det| via blocked (bs=16) in-LDS
// LU with partial pivoting. Panel factorization + triangular solve are
// scalar; the rank-16 Schur complement A22 -= L21*U12 (2/3 of the FLOPs)
// runs on f32 WMMA (complex product = 4 real 16x16x16 matmuls).
// One block per (b,u,v): 16384 blocks x 256 threads.
// ---------------------------------------------------------------------------
__global__ __launch_bounds__(256) void logdet_kernel(
    const float* __restrict__ K,   // [B, C, C, 3, 3]
    float* __restrict__ ld)        // [B]
{
    __shared__ float Ar[64][65];
    __shared__ float Ai[64][65];
    __shared__ float red[64];
    __shared__ int   sPiv;

    const int tid = threadIdx.x;
    const int b   = blockIdx.x >> 12;
    const int f   = blockIdx.x & 4095;
    const int u   = f >> 6;
    const int v   = f & 63;

    // phase factors e^{-2*pi*i*(u*ky + v*kx)/64} per tap
    float cs[TAPS], sn[TAPS];
    #pragma unroll
    for (int t = 0; t < TAPS; ++t) {
        const int ky = t / 3, kx = t % 3;
        const float ang = -2.0f * 3.14159265358979323846f *
                          (float)(u * ky + v * kx) * (1.0f / 64.0f);
        __sincosf(ang, &sn[t], &cs[t]);
    }

    // Build Khat(u,v) = sum_t (K[b,:,:,t] + I*[t==4]) * phase_t  into LDS
    const float* Kb = K + b * CH * CH * TAPS;
    for (int i = tid; i < CH * CH; i += 256) {
        const int co = i >> 6, ci = i & 63;
        const float* Kp = Kb + i * TAPS;
        float sr = 0.f, si = 0.f;
        #pragma unroll
        for (int t = 0; t < TAPS; ++t) {
            float kv = Kp[t];
            if (co == ci && t == 4) kv += 1.0f;
            sr += kv * cs[t];
            si += kv * sn[t];
        }
        Ar[co][ci] = sr;
        Ai[co][ci] = si;
    }
    __syncthreads();

    const int lane  = tid & 31;
    const int wave  = tid >> 5;
    const int lcol  = lane & 15;
    const int lhalf = lane >> 4;
    const v8f zacc  = {0.f,0.f,0.f,0.f,0.f,0.f,0.f,0.f};

    float acc = 0.0f;   // only thread 0's copy is used

    for (int j = 0; j < 4; ++j) {          // block columns of width 16
        const int jb    = j * 16;
        const int jbEnd = jb + 15;

        // ---- panel factorization (columns jb..jb+15, partial pivoting) ----
        for (int k = jb; k <= jbEnd; ++k) {
            if (tid < 64)
                red[tid] = (tid >= k) ? (Ar[tid][k]*Ar[tid][k] + Ai[tid][k]*Ai[tid][k])
                                      : -1.0f;
            __syncthreads();
            if (tid == 0) {
                int p = k; float best = red[k];
                for (int r = k + 1; r < 64; ++r)
                    if (red[r] > best) { best = red[r]; p = r; }
                sPiv = p;
            }
            __syncthreads();
            const int p = sPiv;
            if (p != k) {                     // uniform branch: full-row swap
                if (tid < 64) {
                    float tr = Ar[k][tid]; Ar[k][tid] = Ar[p][tid]; Ar[p][tid] = tr;
                    float ti = Ai[k][tid]; Ai[k][tid] = Ai[p][tid]; Ai[p][tid] = ti;
                }
                __syncthreads();
            }

            const float pr = Ar[k][k], pi = Ai[k][k];
            const float d  = pr * pr + pi * pi;
            if (tid == 0) acc += 0.5f * __logf(d);

            if (k < 63) {
                // multipliers, stored in place (L storage for the WMMA Schur)
                const float invd  = 1.0f / d;
                const int   nrows = 63 - k;
                if (tid < nrows) {
                    const int r = k + 1 + tid;
                    const float ar = Ar[r][k], ai = Ai[r][k];
                    Ar[r][k] = (ar * pr + ai * pi) * invd;
                    Ai[r][k] = (ai * pr - ar * pi) * invd;
                }
                __syncthreads();

                // right-looking update restricted to remaining panel columns
                const int ncols = jbEnd - k;
                if (ncols > 0) {
                    const int total = nrows * ncols;
                    for (int idx = tid; idx < total; idx += 256) {
                        const int r = k + 1 + idx / ncols;
                        const int c = k + 1 + idx % ncols;
                        const float ur = Ar[k][c], ui = Ai[k][c];
                        const float mr = Ar[r][k], mi = Ai[r][k];
                        Ar[r][c] -= mr * ur - mi * ui;
                        Ai[r][c] -= mr * ui + mi * ur;
                    }
                }
                __syncthreads();
            }
        }

        if (j == 3) break;
        const int cbase0 = jb + 16;
        const int ntc    = 64 - cbase0;     // trailing columns: 48, 32, 16

        // ---- triangular solve U12 = L11^{-1} * A12 (one column per thread,
        //      barrier-free: column c depends only on column c) ----
        if (tid < ntc) {
            const int c = cbase0 + tid;
            float ur[16], ui[16];
            #pragma unroll
            for (int i = 0; i < 16; ++i) {
                float vr = Ar[jb + i][c], vi = Ai[jb + i][c];
                #pragma unroll
                for (int m = 0; m < i; ++m) {
                    const float lr = Ar[jb + i][jb + m], li = Ai[jb + i][jb + m];
                    vr -= lr * ur[m] - li * ui[m];
                    vi -= lr * ui[m] + li * ur[m];
                }
                ur[i] = vr; ui[i] = vi;
                Ar[jb + i][c] = vr;
                Ai[jb + i][c] = vi;
            }
        }
        __syncthreads();

        // ---- Schur complement A22 -= L21 * U12 via f32 WMMA ----
        // complex 16x16 (K=16) tile = 4 real matmuls = 16 wmma_f32_16x16x4.
        const int nt     = 3 - j;            // trailing 16-tiles per dimension
        const int ntiles = nt * nt;
        for (int tt = wave; tt < ntiles; tt += 8) {   // wave-uniform => EXEC all 1s
            const int rbase = cbase0 + (tt / nt) * 16;
            const int cb    = cbase0 + (tt % nt) * 16;
            v8f accR = zacc, accI = zacc;
            #pragma unroll
            for (int kc = 0; kc < 4; ++kc) {
                const int k0   = jb + kc * 4 + lhalf * 2;  // K indices {k0, k0+1}
                const int arow = rbase + lcol;
                v2f lr, li, nli, urf, uif;
                lr.x  = Ar[arow][k0];      lr.y  = Ar[arow][k0 + 1];
                li.x  = Ai[arow][k0];      li.y  = Ai[arow][k0 + 1];
                urf.x = Ar[k0][cb + lcol]; urf.y = Ar[k0 + 1][cb + lcol];
                uif.x = Ai[k0][cb + lcol]; uif.y = Ai[k0 + 1][cb + lcol];
                nli.x = -li.x;             nli.y = -li.y;
                accR = WMMA_F32X4(lr,  urf, accR);   // + Lr*Ur
                accR = WMMA_F32X4(nli, uif, accR);   // - Li*Ui
                accI = WMMA_F32X4(lr,  uif, accI);   // + Lr*Ui
                accI = WMMA_F32X4(li,  urf, accI);   // + Li*Ur
            }
            #pragma unroll
            for (int r = 0; r < 8; ++r) {
                const int row = rbase + r + 8 * lhalf;
                const int col = cb + lcol;
                Ar[row][col] -= accR[r];
                Ai[row][col] -= accI[r];
            }
        }
        __syncthreads();
    }

    if (tid == 0) atomicAdd(ld + b, acc);
}

__global__ void zero_logdet_kernel(float* __restrict__ p)
{
    if (threadIdx.x < BATCH) p[threadIdx.x] = 0.0f;
}

extern "C" void kernel_launch(void* const* d_in, const int* in_sizes, int n_in,
                              void* d_out, int out_size, void* d_ws, size_t ws_size,
                              hipStream_t stream)
{
    const float* X    = (const float*)d_in[0];  // conv_in [4,64,64,64]
    const float* K    = (const float*)d_in[1];  // K       [4,64,64,3,3]
    const float* bias = (const float*)d_in[2];  // bias    [4,64,1,1]
    float* out = (float*)d_out;                 // conv_out ‖ logdet
    float* ld  = out + BATCH * CH * SPAT;       // + 1048576

    zero_logdet_kernel<<<1, 32, 0, stream>>>(ld);
    conv_wmma_kernel<<<128, 256, 0, stream>>>(X, K, bias, out);
    logdet_kernel<<<BATCH * SPAT, 256, 0, stream>>>(K, ld);
}